// ParallelFullyConnectedNN_20624432955530
// MI455X (gfx1250) — compile-verified
//
#include <hip/hip_runtime.h>
#include <hip/hip_bf16.h>

// ---------------------------------------------------------------------------
// Fused 64-way parallel SIREN evaluation for MI455X (gfx1250, wave32, WMMA).
//   h = sin(30*(x@W0+b0)); 4x h = sin(30*(h@Wi+bi)); out = h@Wout+bout
// Hidden GEMMs on V_WMMA_F32_16X16X4_F32 (f32 — sin(30*z) is phase-sensitive).
// Activations live in LDS; weights staged per layer into LDS, k-quad
// interleaved so each B fragment is one conflict-free ds_load_b64.
// ---------------------------------------------------------------------------

typedef __attribute__((ext_vector_type(2))) float v2f;
typedef __attribute__((ext_vector_type(8))) float v8f;

#define NPTS   16384
#define KNET   64
#define IN_F   3
#define HID    128
#define FREQ   30.0f
#define MTILE  64          // points per block
#define HSTR   132         // padded LDS row stride (dwords): rows 4 banks apart
#define NHID   4

#define SMEM_FLOATS (MTILE * HSTR + HID * HID)
#define SMEM_BYTES  (SMEM_FLOATS * 4)

__global__ __launch_bounds__(256)
void siren_fused_kernel(const float* __restrict__ x,
                        const float* __restrict__ W0,  const float* __restrict__ b0,
                        const float* __restrict__ W1,  const float* __restrict__ b1,
                        const float* __restrict__ W2,  const float* __restrict__ b2,
                        const float* __restrict__ W3,  const float* __restrict__ b3,
                        const float* __restrict__ W4,  const float* __restrict__ b4,
                        const float* __restrict__ Wout, const float* __restrict__ bout,
                        float* __restrict__ out)
{
    extern __shared__ float smem[];
    float* hs = smem;                    // [MTILE][HSTR] activation tile
    float* ws = smem + MTILE * HSTR;     // [32][HID][4]  k-quad-interleaved W

    const int tid  = threadIdx.x;
    const int lane = tid & 31;
    const int wave = tid >> 5;
    const int n0   = blockIdx.x * MTILE; // point-tile base
    const int k    = blockIdx.y;         // network id

    const float* Ws[NHID] = {W1, W2, W3, W4};
    const float* Bs[NHID] = {b1, b2, b3, b4};

    // ---------------- Layer 0: [64 x 3] @ [3 x 128] + b, sin ----------------
    // Stage the 64x3 x-tile into (currently unused) ws.
    if (tid < MTILE * IN_F) ws[tid] = x[(size_t)n0 * IN_F + tid];
    __syncthreads();
    {
        const int j     = tid & (HID - 1);   // column, fixed per thread
        const int rbase = (tid >> 7) * 32;   // rows 0..31 or 32..63
        const size_t w0b = (size_t)k * IN_F * HID;
        const float wa = W0[w0b + 0 * HID + j];
        const float wb = W0[w0b + 1 * HID + j];
        const float wc = W0[w0b + 2 * HID + j];
        const float bj = b0[k * HID + j];
        #pragma unroll 4
        for (int r = rbase; r < rbase + 32; ++r) {
            const float* xr = ws + r * IN_F;
            float s = fmaf(xr[0], wa, fmaf(xr[1], wb, fmaf(xr[2], wc, bj)));
            hs[r * HSTR + j] = __sinf(FREQ * s);
        }
    }
    __syncthreads();   // layer-0 reads of ws done; hs tile complete

    // ---------------- 4 hidden layers on f32 WMMA ----------------
    // Wave w -> 16-row x 64-col patch: rows r0 = (w&3)*16, cols cg = (w>>2)*64.
    const int r0  = (wave & 3) * 16;
    const int cg  = (wave >> 2) * 64;
    const int c15 = lane & 15;
    const int hi  = lane >> 4;           // lane half: 0 or 1

    for (int layer = 0; layer < NHID; ++layer) {
        // Stage W[layer] into LDS, permuted to [k4][col][4] (k4 = kk>>2).
        // Each thread: 4 coalesced global b32 (one column over a k-quad) and
        // one ds_store_b128 at col*16B -> wave spans all 64 banks twice (min).
        {
            const float* src = Ws[layer] + (size_t)k * HID * HID;
            #pragma unroll
            for (int i = 0; i < (HID * HID / 4) / 256; ++i) {   // 16 quads/thr
                const int p   = tid + i * 256;
                const int k4  = p >> 7;              // 0..31
                const int col = p & (HID - 1);       // 0..127
                const float* g = src + (k4 * 4) * HID + col;
                float4 q;
                q.x = g[0 * HID];
                q.y = g[1 * HID];
                q.z = g[2 * HID];
                q.w = g[3 * HID];
                *(float4*)&ws[k4 * 4 * HID + col * 4] = q;
            }
        }
        // CDNA5 prefetch path: pull the NEXT layer's weights (or Wout) toward
        // this WGP while the K-loop below runs (global_prefetch_b8).
        if (layer + 1 < NHID) {
            const float* nxt = Ws[layer + 1] + (size_t)k * HID * HID;
            __builtin_prefetch(nxt + tid * 64, 0, 1);        // 256 thr x 256B
            __builtin_prefetch(nxt + tid * 64 + 32, 0, 1);
        } else if (wave == 0) {
            __builtin_prefetch(Wout + (size_t)k * HID + lane * 4, 0, 1);
        }
        __syncthreads();  // ws ready; hs writes from previous layer visible

        // Seed accumulators with bias (bias depends only on column).
        const float* bl = Bs[layer] + k * HID;
        v8f acc[4];
        #pragma unroll
        for (int t = 0; t < 4; ++t) {
            const float bb = bl[cg + t * 16 + c15];
            acc[t] = (v8f){bb, bb, bb, bb, bb, bb, bb, bb};
        }

        // K loop: 128 in chunks of 4; one b64 A-frag + four b64 B-frags per
        // step, all conflict-free, feeding 4 WMMAs.
        #pragma unroll 8
        for (int kk = 0; kk < HID; kk += 4) {
            // A 16x4 frag: lanes 0-15 rows r0+lane {K=kk,kk+1};
            //              lanes 16-31 same rows  {K=kk+2,kk+3}.
            const int ak = kk + hi * 2;
            v2f a = *(const v2f*)&hs[(r0 + c15) * HSTR + ak];
            // B 4x16 frags from k-quad layout: {W[ak][col], W[ak+1][col]}.
            const float* wq = ws + (kk >> 2) * 4 * HID + (cg + c15) * 4 + hi * 2;
            #pragma unroll
            for (int t = 0; t < 4; ++t) {
                v2f b = *(const v2f*)(wq + t * 64);
                acc[t] = __builtin_amdgcn_wmma_f32_16x16x4_f32(
                    /*neg_a=*/false, a, /*neg_b=*/false, b,
                    /*c_mod=*/(short)0, acc[t],
                    /*reuse_a=*/false, /*reuse_b=*/false);
            }
        }

        __syncthreads();  // all hs reads for this layer complete

        // Epilogue: sin(FREQ*acc) -> hs. C/D layout: VGPR v, lanes 0-15 row
        // r0+v, lanes 16-31 row r0+v+8, col = cg + t*16 + (lane&15).
        #pragma unroll
        for (int t = 0; t < 4; ++t) {
            const int col = cg + t * 16 + c15;
            #pragma unroll
            for (int v = 0; v < 8; ++v) {
                hs[(r0 + v + hi * 8) * HSTR + col] = __sinf(FREQ * acc[t][v]);
            }
        }
        __syncthreads();  // hs ready before next stage/read
    }

    // ---------------- Output layer: [64 x 128] @ [128 x 1] + bout ----------
    {
        const float* wo  = Wout + (size_t)k * HID;
        const int row = tid >> 2;     // 0..63
        const int q   = tid & 3;      // quarter of K
        const float* hr = hs + row * HSTR + q * 32;
        const float* wq = wo + q * 32;
        float s = 0.0f;
        #pragma unroll
        for (int i = 0; i < 32; ++i) s = fmaf(hr[i], wq[i], s);
        s += __shfl_xor(s, 1);
        s += __shfl_xor(s, 2);
        if (q == 0) out[(size_t)k * NPTS + n0 + row] = s + bout[k];
    }
}

extern "C" void kernel_launch(void* const* d_in, const int* in_sizes, int n_in,
                              void* d_out, int out_size, void* d_ws, size_t ws_size,
                              hipStream_t stream) {
    const float* x    = (const float*)d_in[0];
    const float* W0   = (const float*)d_in[1];
    const float* b0   = (const float*)d_in[2];
    const float* W1   = (const float*)d_in[3];
    const float* b1   = (const float*)d_in[4];
    const float* W2   = (const float*)d_in[5];
    const float* b2   = (const float*)d_in[6];
    const float* W3   = (const float*)d_in[7];
    const float* b3   = (const float*)d_in[8];
    const float* W4   = (const float*)d_in[9];
    const float* b4   = (const float*)d_in[10];
    const float* Wout = (const float*)d_in[11];
    const float* bout = (const float*)d_in[12];
    float* out = (float*)d_out;

    // ~97 KB dynamic LDS (gfx1250 WGP has 320 KB); raise the attribute cap.
    (void)hipFuncSetAttribute((const void*)siren_fused_kernel,
                              hipFuncAttributeMaxDynamicSharedMemorySize,
                              SMEM_BYTES);

    dim3 grid(NPTS / MTILE, KNET);
    dim3 block(256);
    siren_fused_kernel<<<grid, block, SMEM_BYTES, stream>>>(
        x, W0, b0, W1, b1, W2, b2, W3, b3, W4, b4, Wout, bout, out);
}